// VariationalGCNEncoder_21543555956945
// MI455X (gfx1250) — compile-verified
//
#include <hip/hip_runtime.h>
#include <hip/hip_bf16.h>
#include <math.h>

typedef __attribute__((ext_vector_type(2))) float v2f;
typedef __attribute__((ext_vector_type(8))) float v8f;

// ---------------- utility kernels ----------------

__global__ void zero_f32(float* __restrict__ p, long long n) {
  long long i = (long long)blockIdx.x * blockDim.x + threadIdx.x;
  long long stride = (long long)gridDim.x * blockDim.x;
  for (; i < n; i += stride) p[i] = 0.0f;
}

__global__ void deg_count(const int* __restrict__ dst, float* __restrict__ deg, int E) {
  int e = blockIdx.x * blockDim.x + threadIdx.x;
  if (e < E) atomicAdd(&deg[dst[e]], 1.0f);
}

__global__ void deg_to_dinv(float* __restrict__ deg, int N) {
  int i = blockIdx.x * blockDim.x + threadIdx.x;
  if (i < N) {
    float d = deg[i];
    deg[i] = (d > 0.0f) ? rsqrtf(fmaxf(d, 1.0f)) : 0.0f;
  }
}

__global__ void edge_norm(const int* __restrict__ src, const int* __restrict__ dst,
                          const float* __restrict__ dinv, float* __restrict__ norm, int E) {
  int e = blockIdx.x * blockDim.x + threadIdx.x;
  if (e < E) norm[e] = dinv[src[e]] * dinv[dst[e]];
}

// Pack w_mu[128,64] and w_ls[128,64] into one [128,128] matrix (cols 0-63 = mu).
__global__ void pack_w(const float* __restrict__ w_mu, const float* __restrict__ w_ls,
                       float* __restrict__ wcat) {
  int idx = blockIdx.x * blockDim.x + threadIdx.x;
  if (idx < 128 * 128) {
    int k = idx >> 7, j = idx & 127;
    wcat[idx] = (j < 64) ? w_mu[k * 64 + j] : w_ls[k * 64 + (j - 64)];
  }
}

// ---------------- WMMA fp32 GEMM: C[M,128] = A[M,128] @ B[128,128] ----------------
// One wave per 16x16 output tile, V_WMMA_F32_16X16X4_F32, K-loop of 32 steps.
// Block = 256 threads = 8 waves = one M-stripe covering all 8 N-tiles.
__global__ __launch_bounds__(256) void gemm_f32_wmma(
    const float* __restrict__ A, const float* __restrict__ B,
    float* __restrict__ C, int M) {
  const int K = 128, N = 128;
  const int wave  = threadIdx.x >> 5;
  const int lane  = threadIdx.x & 31;
  const int half  = lane >> 4;   // 0: K pair {0,1} / M rows 0-7 ; 1: K pair {2,3} / M rows 8-15
  const int l16   = lane & 15;
  const int tileM = blockIdx.x * 16;
  const int tileN = wave * 16;

  int rowA = tileM + l16;
  if (rowA >= M) rowA = M - 1;                 // branchless clamp keeps EXEC all-ones
  const float* arow = A + (size_t)rowA * K;
  const float* bcol = B + (tileN + l16);

  v8f acc = {};
#pragma unroll 8
  for (int kk = 0; kk < K; kk += 4) {
    // A 16x4 layout: lane<16 -> K=kk+{0,1}; lane>=16 -> K=kk+{2,3} (8B-aligned pair)
    v2f a = *(const v2f*)(arow + kk + 2 * half);
    // B 4x16 layout: VGPR0 = row K=kk+2*half, VGPR1 = row K=kk+2*half+1, N = tileN+l16
    v2f b;
    b.x = bcol[(size_t)(kk + 2 * half) * N];
    b.y = bcol[(size_t)(kk + 2 * half + 1) * N];
    acc = __builtin_amdgcn_wmma_f32_16x16x4_f32(
        /*neg_a=*/false, a, /*neg_b=*/false, b,
        /*c_mod=*/(short)0, acc, /*reuse_a=*/false, /*reuse_b=*/false);
  }

  // C/D layout: VGPR r -> M = tileM + r + 8*half, N = tileN + l16
  float* crow = C + (size_t)(tileM + half * 8) * N + tileN + l16;
  if (tileM + 16 <= M) {
    // Full tile (always the case for M % 16 == 0): straight-line coalesced stores.
#pragma unroll
    for (int r = 0; r < 8; ++r) crow[(size_t)r * N] = acc[r];
  } else {
#pragma unroll
    for (int r = 0; r < 8; ++r) {
      if (tileM + half * 8 + r < M) crow[(size_t)r * N] = acc[r];
    }
  }
}

// ---------------- edge gather * norm -> atomic scatter (D = 128) ----------------
// 32 lanes per edge, 4 contiguous floats per lane: fully coalesced 512B row reads.
__global__ void scatter_edges(const float* __restrict__ hsrc,
                              const int* __restrict__ src, const int* __restrict__ dst,
                              const float* __restrict__ norm,
                              float* __restrict__ agg, int E) {
  long long t = (long long)blockIdx.x * blockDim.x + threadIdx.x;
  long long total = (long long)E * 32;
  if (t >= total) return;
  int e  = (int)(t >> 5);
  int c4 = ((int)t & 31) << 2;
  float w = norm[e];
  const float4 v = *(const float4*)(hsrc + (size_t)src[e] * 128 + c4);
  float* o = agg + (size_t)dst[e] * 128 + c4;
  atomicAdd(o + 0, v.x * w);
  atomicAdd(o + 1, v.y * w);
  atomicAdd(o + 2, v.z * w);
  atomicAdd(o + 3, v.w * w);
}

__global__ void bias_relu(float* __restrict__ h, const float* __restrict__ b, long long total) {
  long long i = (long long)blockIdx.x * blockDim.x + threadIdx.x;
  if (i < total) {
    int c = (int)(i & 127);
    float v = h[i] + b[c];
    h[i] = v > 0.0f ? v : 0.0f;
  }
}

// agg2[N,128] -> out = [mu (N*64) | logstd (N*64)] with biases
__global__ void finalize_out(const float* __restrict__ agg2,
                             const float* __restrict__ b_mu, const float* __restrict__ b_ls,
                             float* __restrict__ out, int Nn) {
  long long i = (long long)blockIdx.x * blockDim.x + threadIdx.x;
  long long total = (long long)Nn * 128;
  if (i < total) {
    int node = (int)(i >> 7);
    int c    = (int)(i & 127);
    float v  = agg2[i];
    if (c < 64) out[(size_t)node * 64 + c] = v + b_mu[c];
    else        out[(size_t)Nn * 64 + (size_t)node * 64 + (c - 64)] = v + b_ls[c - 64];
  }
}

// ---------------- launch ----------------

extern "C" void kernel_launch(void* const* d_in, const int* in_sizes, int n_in,
                              void* d_out, int out_size, void* d_ws, size_t ws_size,
                              hipStream_t stream) {
  const float* x    = (const float*)d_in[0];
  const int*   ei   = (const int*)d_in[1];
  const float* w1   = (const float*)d_in[2];
  const float* b1   = (const float*)d_in[3];
  const float* w_mu = (const float*)d_in[4];
  const float* b_mu = (const float*)d_in[5];
  const float* w_ls = (const float*)d_in[6];
  const float* b_ls = (const float*)d_in[7];

  const int Nn = in_sizes[0] / 128;      // 50000
  const int E  = in_sizes[1] / 2;        // 800000
  const int* src = ei;
  const int* dst = ei + E;

  char* ws = (char*)d_ws;
  size_t off = 0;
  auto alloc = [&](size_t bytes) -> void* {
    void* p = ws + off;
    off = (off + bytes + 255) & ~(size_t)255;
    return p;
  };
  float* dinv = (float*)alloc((size_t)Nn * 4);            // degree, then deg^-1/2 in place
  float* norm = (float*)alloc((size_t)E * 4);             // per-edge norm
  float* hlin = (float*)alloc((size_t)Nn * 128 * 4);      // GEMM output (reused for layer 2)
  float* agg1 = (float*)alloc((size_t)Nn * 128 * 4);      // layer-1 aggregate -> h
  float* agg2 = (float*)alloc((size_t)Nn * 128 * 4);      // layer-2 aggregate
  float* wcat = (float*)alloc((size_t)128 * 128 * 4);     // [w_mu | w_ls]
  (void)ws_size; (void)n_in; (void)out_size;

  const int B = 256;
  const long long tot = (long long)Nn * 128;
  const long long st  = (long long)E * 32;

  // normalization factors (recomputed every call: deterministic)
  zero_f32<<<1024, B, 0, stream>>>(dinv, Nn);
  deg_count<<<(E + B - 1) / B, B, 0, stream>>>(dst, dinv, E);
  deg_to_dinv<<<(Nn + B - 1) / B, B, 0, stream>>>(dinv, Nn);
  edge_norm<<<(E + B - 1) / B, B, 0, stream>>>(src, dst, dinv, norm, E);
  pack_w<<<(128 * 128 + B - 1) / B, B, 0, stream>>>(w_mu, w_ls, wcat);

  // ----- layer 1: h = relu(agg(x @ w1) + b1) -----
  gemm_f32_wmma<<<(Nn + 15) / 16, B, 0, stream>>>(x, w1, hlin, Nn);
  zero_f32<<<2048, B, 0, stream>>>(agg1, tot);
  scatter_edges<<<(int)((st + B - 1) / B), B, 0, stream>>>(hlin, src, dst, norm, agg1, E);
  bias_relu<<<(int)((tot + B - 1) / B), B, 0, stream>>>(agg1, b1, tot);

  // ----- layer 2: mu / logstd via fused [128,128] GEMM -----
  gemm_f32_wmma<<<(Nn + 15) / 16, B, 0, stream>>>(agg1, wcat, hlin, Nn);
  zero_f32<<<2048, B, 0, stream>>>(agg2, tot);
  scatter_edges<<<(int)((st + B - 1) / B), B, 0, stream>>>(hlin, src, dst, norm, agg2, E);
  finalize_out<<<(int)((tot + B - 1) / B), B, 0, stream>>>(agg2, b_mu, b_ls, (float*)d_out, Nn);
}